// RBFInterpolator_19095424598221
// MI455X (gfx1250) — compile-verified
//
#include <hip/hip_runtime.h>
#include <hip/hip_bf16.h>
#include <math.h>

// RBF thin-plate-spline interpolator for MI455X (gfx1250).
// fp32 path via V_WMMA_F32_16X16X4_F32 for: LU trailing updates, triangular
// solves (via explicit 16x16 block inverses), and the fused Kx@coeffs GEMM.

typedef __attribute__((ext_vector_type(2))) float v2f;
typedef __attribute__((ext_vector_type(8))) float v8f;

#define P_N   4096   // y points
#define Q_N   8192   // x points
#define S_N   64     // rhs columns
#define R_N   4      // poly terms (degree 1, ndim 3)
#define NP_N  4100   // P + R
#define NN    4112   // padded system size (257*16)
#define K16   257    // NN / 16

// ---------------- device helpers ----------------

__device__ __forceinline__ float tps(float sq) {
  return (sq > 0.f) ? 0.5f * sq * __logf(sq) : 0.f;
}

// A operand: 16x4 f32, 2 VGPRs. lane<16: K=kb+0/1 ; lane>=16: K=kb+2/3
__device__ __forceinline__ v2f load_a(const float* A, int lda, int kb, int lane) {
  int row = lane & 15;
  int kk  = kb + ((lane >> 4) << 1);
  v2f a;
  a.x = A[(size_t)row * lda + kk];
  a.y = A[(size_t)row * lda + kk + 1];
  return a;
}
__device__ __forceinline__ v2f load_a_neg(const float* A, int lda, int kb, int lane) {
  int row = lane & 15;
  int kk  = kb + ((lane >> 4) << 1);
  v2f a;
  a.x = -A[(size_t)row * lda + kk];
  a.y = -A[(size_t)row * lda + kk + 1];
  return a;
}
// B operand: 4x16 f32, 2 VGPRs. lane<16: rows kb+0/1 ; lane>=16: rows kb+2/3
__device__ __forceinline__ v2f load_b(const float* B, int ldb, int kb, int lane) {
  int col = lane & 15;
  int kk  = kb + ((lane >> 4) << 1);
  v2f b;
  b.x = B[(size_t)kk * ldb + col];
  b.y = B[(size_t)(kk + 1) * ldb + col];
  return b;
}
// C/D: 16x16 f32, 8 VGPRs. VGPR v: row = v + 8*(lane>=16), col = lane&15
__device__ __forceinline__ v8f load_c(const float* C, int ldc, int lane) {
  int col = lane & 15;
  int rb  = (lane >> 4) << 3;
  v8f c;
#pragma unroll
  for (int v = 0; v < 8; ++v) c[v] = C[(size_t)(rb + v) * ldc + col];
  return c;
}
__device__ __forceinline__ void store_c(float* C, int ldc, int lane, v8f c) {
  int col = lane & 15;
  int rb  = (lane >> 4) << 3;
#pragma unroll
  for (int v = 0; v < 8; ++v) C[(size_t)(rb + v) * ldc + col] = c[v];
}

// ---------------- setup kernels ----------------

__global__ void stats_kernel(const float* __restrict__ y, float* __restrict__ stats) {
  __shared__ float smn[3][256];
  __shared__ float smx[3][256];
  int tid = threadIdx.x;
  float mn0 = 1e30f, mn1 = 1e30f, mn2 = 1e30f;
  float mx0 = -1e30f, mx1 = -1e30f, mx2 = -1e30f;
  for (int r = tid; r < P_N; r += 256) {
    float a = y[r * 3 + 0], b = y[r * 3 + 1], c = y[r * 3 + 2];
    mn0 = fminf(mn0, a); mx0 = fmaxf(mx0, a);
    mn1 = fminf(mn1, b); mx1 = fmaxf(mx1, b);
    mn2 = fminf(mn2, c); mx2 = fmaxf(mx2, c);
  }
  smn[0][tid] = mn0; smn[1][tid] = mn1; smn[2][tid] = mn2;
  smx[0][tid] = mx0; smx[1][tid] = mx1; smx[2][tid] = mx2;
  __syncthreads();
  for (int s = 128; s > 0; s >>= 1) {
    if (tid < s) {
#pragma unroll
      for (int d = 0; d < 3; ++d) {
        smn[d][tid] = fminf(smn[d][tid], smn[d][tid + s]);
        smx[d][tid] = fmaxf(smx[d][tid], smx[d][tid + s]);
      }
    }
    __syncthreads();
  }
  if (tid < 3) {
    float mn = smn[tid][0], mx = smx[tid][0];
    float shift = 0.5f * (mx + mn);
    float scale = 0.5f * (mx - mn);
    if (scale == 0.f) scale = 1.f;
    stats[tid] = shift;
    stats[3 + tid] = scale;
  }
}

__global__ void build_lhs(const float* __restrict__ y, const float* __restrict__ stats,
                          float* __restrict__ lhs) {
  int tid = threadIdx.x;
  int i = blockIdx.y * 16 + (tid >> 4);
  int j = blockIdx.x * 16 + (tid & 15);
  float v;
  if (i < P_N && j < P_N) {
    float dx = y[i * 3 + 0] - y[j * 3 + 0];
    float dy = y[i * 3 + 1] - y[j * 3 + 1];
    float dz = y[i * 3 + 2] - y[j * 3 + 2];
    v = tps(dx * dx + dy * dy + dz * dz);
  } else if (i < P_N) {
    int r = j - P_N;
    v = (r == 0) ? 1.f : ((r < R_N) ? (y[i * 3 + r - 1] - stats[r - 1]) / stats[3 + r - 1] : 0.f);
  } else if (j < P_N) {
    int r = i - P_N;
    v = (r == 0) ? 1.f : ((r < R_N) ? (y[j * 3 + r - 1] - stats[r - 1]) / stats[3 + r - 1] : 0.f);
  } else {
    v = (i >= NP_N || j >= NP_N) ? ((i == j) ? 1.f : 0.f) : 0.f;
  }
  lhs[(size_t)i * NN + j] = v;
}

__global__ void build_rhs(const float* __restrict__ d, float* __restrict__ rhs) {
  int idx = blockIdx.x * 256 + threadIdx.x;
  if (idx >= NN * S_N) return;
  int row = idx / S_N;
  int col = idx - row * S_N;
  rhs[idx] = (row < P_N) ? d[row * S_N + col] : 0.f;
}

// ---------------- LU with partial pivoting ----------------

// One workgroup: factor 16-wide panel (rows k*16..NN) with partial pivoting,
// apply the 16 row swaps to all columns outside the panel, and compute
// inv(L_kk) / inv(U_kk) into invLU (k*512 : +256 = invL, +256.. = invU).
__global__ void panel_factor(float* __restrict__ lhs, int* __restrict__ ipiv,
                             float* __restrict__ invLU, int k) {
  __shared__ float sv[1024];
  __shared__ int   si[1024];
  __shared__ float urow[16];
  __shared__ int   spiv[16];
  __shared__ float sdiag[256];
  int tid = threadIdx.x;
  int k16 = k * 16;
  for (int c = 0; c < 16; ++c) {
    int jc = k16 + c;
    int r0 = k16 + c;
    // argmax |A[r][jc]| over trailing column
    float best = -1.f;
    int bidx = r0;
    for (int r = r0 + tid; r < NN; r += 1024) {
      float v = fabsf(lhs[(size_t)r * NN + jc]);
      if (v > best) { best = v; bidx = r; }
    }
    sv[tid] = best; si[tid] = bidx;
    __syncthreads();
    for (int s = 512; s > 0; s >>= 1) {
      if (tid < s && sv[tid + s] > sv[tid]) { sv[tid] = sv[tid + s]; si[tid] = si[tid + s]; }
      __syncthreads();
    }
    int prow = si[0];
    if (tid == 0) { ipiv[jc] = prow; spiv[c] = prow; }
    // swap panel-width row segments
    if (prow != r0 && tid < 16) {
      float t = lhs[(size_t)r0 * NN + k16 + tid];
      lhs[(size_t)r0 * NN + k16 + tid] = lhs[(size_t)prow * NN + k16 + tid];
      lhs[(size_t)prow * NN + k16 + tid] = t;
    }
    __syncthreads();
    if (tid < 16) urow[tid] = lhs[(size_t)r0 * NN + k16 + tid];
    __syncthreads();
    float piv = urow[c];
    float inv = (piv != 0.f) ? 1.f / piv : 0.f;
    // scale column + rank-1 update of remaining panel columns
    for (int r = r0 + 1 + tid; r < NN; r += 1024) {
      float lval = lhs[(size_t)r * NN + jc] * inv;
      lhs[(size_t)r * NN + jc] = lval;
      for (int cc = c + 1; cc < 16; ++cc)
        lhs[(size_t)r * NN + k16 + cc] -= lval * urow[cc];
    }
    __syncthreads();
  }
  // ---- apply recorded swaps to all columns outside the panel ----
  for (int t = tid; t < NN - 16; t += 1024) {
    int j = (t < k16) ? t : t + 16;
    for (int i = 0; i < 16; ++i) {
      int r = k16 + i, p = spiv[i];
      if (p != r) {
        float tmp = lhs[(size_t)r * NN + j];
        lhs[(size_t)r * NN + j] = lhs[(size_t)p * NN + j];
        lhs[(size_t)p * NN + j] = tmp;
      }
    }
  }
  // ---- invert the 16x16 diagonal block (unit-lower L, upper U) ----
  if (tid < 256) sdiag[tid] = lhs[(size_t)(k16 + (tid >> 4)) * NN + k16 + (tid & 15)];
  __syncthreads();
  if (tid < 16) {                 // invL column tid
    int c = tid;
    float x[16];
#pragma unroll
    for (int r = 0; r < 16; ++r) {
      float v = (r == c) ? 1.f : 0.f;
#pragma unroll
      for (int m = 0; m < 16; ++m)
        if (m < r) v -= sdiag[r * 16 + m] * x[m];
      x[r] = (r < c) ? 0.f : v;
    }
#pragma unroll
    for (int r = 0; r < 16; ++r) invLU[(size_t)k * 512 + r * 16 + c] = x[r];
  } else if (tid < 32) {          // invU column tid-16
    int c = tid - 16;
    float x[16];
#pragma unroll
    for (int r = 15; r >= 0; --r) {
      float v = (r == c) ? 1.f : 0.f;
#pragma unroll
      for (int m = 0; m < 16; ++m)
        if (m > r) v -= sdiag[r * 16 + m] * x[m];
      float dg = sdiag[r * 16 + r];
      x[r] = (r > c) ? 0.f : ((dg != 0.f) ? v / dg : 0.f);
    }
#pragma unroll
    for (int r = 0; r < 16; ++r) invLU[(size_t)k * 512 + 256 + r * 16 + c] = x[r];
  }
}

// TRSM as WMMA: U[k][j] = invL_kk @ A[k][j]; one wave per tile.
__global__ void row_trsm(float* __restrict__ lhs, const float* __restrict__ invLU, int k) {
  int lane = threadIdx.x;
  int j = k + 1 + blockIdx.x;
  const float* invL = invLU + (size_t)k * 512;
  float* B = lhs + (size_t)(k * 16) * NN + j * 16;
  v8f c;
#pragma unroll
  for (int v = 0; v < 8; ++v) c[v] = 0.f;
#pragma unroll
  for (int kb = 0; kb < 16; kb += 4) {
    v2f a = load_a(invL, 16, kb, lane);
    v2f b = load_b(B, NN, kb, lane);
    c = __builtin_amdgcn_wmma_f32_16x16x4_f32(false, a, false, b, (short)0, c, false, false);
  }
  store_c(B, NN, lane, c);
}

// Trailing update: 32x32 per wave (2x2 register tiles, shared A/B fragments).
__global__ void lu_gemm(float* __restrict__ lhs, int k) {
  int lane = threadIdx.x;
  int i0 = k + 1 + (int)blockIdx.y * 2;
  int j0 = k + 1 + (int)blockIdx.x * 2;
  int ni = (i0 + 1 < K16) ? 2 : 1;
  int nj = (j0 + 1 < K16) ? 2 : 1;
  const float* A0 = lhs + (size_t)(i0 * 16) * NN + k * 16;
  const float* B0 = lhs + (size_t)(k * 16) * NN + j0 * 16;
  float*       C0 = lhs + (size_t)(i0 * 16) * NN + j0 * 16;
  v8f c00, c01, c10, c11;
#pragma unroll
  for (int v = 0; v < 8; ++v) { c00[v] = 0.f; c01[v] = 0.f; c10[v] = 0.f; c11[v] = 0.f; }
  c00 = load_c(C0, NN, lane);
  if (nj > 1) c01 = load_c(C0 + 16, NN, lane);
  if (ni > 1) c10 = load_c(C0 + (size_t)16 * NN, NN, lane);
  if (ni > 1 && nj > 1) c11 = load_c(C0 + (size_t)16 * NN + 16, NN, lane);
#pragma unroll
  for (int kb = 0; kb < 16; kb += 4) {
    v2f a0 = load_a_neg(A0, NN, kb, lane);
    v2f b0 = load_b(B0, NN, kb, lane);
    c00 = __builtin_amdgcn_wmma_f32_16x16x4_f32(false, a0, false, b0, (short)0, c00, false, false);
    if (nj > 1) {
      v2f b1 = load_b(B0 + 16, NN, kb, lane);
      c01 = __builtin_amdgcn_wmma_f32_16x16x4_f32(false, a0, false, b1, (short)0, c01, false, false);
      if (ni > 1) {
        v2f a1 = load_a_neg(A0 + (size_t)16 * NN, NN, kb, lane);
        c10 = __builtin_amdgcn_wmma_f32_16x16x4_f32(false, a1, false, b0, (short)0, c10, false, false);
        c11 = __builtin_amdgcn_wmma_f32_16x16x4_f32(false, a1, false, b1, (short)0, c11, false, false);
      }
    } else if (ni > 1) {
      v2f a1 = load_a_neg(A0 + (size_t)16 * NN, NN, kb, lane);
      c10 = __builtin_amdgcn_wmma_f32_16x16x4_f32(false, a1, false, b0, (short)0, c10, false, false);
    }
  }
  store_c(C0, NN, lane, c00);
  if (nj > 1) store_c(C0 + 16, NN, lane, c01);
  if (ni > 1) store_c(C0 + (size_t)16 * NN, NN, lane, c10);
  if (ni > 1 && nj > 1) store_c(C0 + (size_t)16 * NN + 16, NN, lane, c11);
}

// ---------------- triangular solves on RHS (fused, 1 launch/step) ----------------

__global__ void laswp_rhs(float* __restrict__ rhs, const int* __restrict__ ipiv) {
  int s = threadIdx.x;  // 64 columns
  for (int r = 0; r < NN; ++r) {
    int p = ipiv[r];
    if (p != r) {
      float t = rhs[(size_t)r * S_N + s];
      rhs[(size_t)r * S_N + s] = rhs[(size_t)p * S_N + s];
      rhs[(size_t)p * S_N + s] = t;
    }
  }
}

// Forward: every block computes W = invL_kk @ rhs[k] (into LDS); blockIdx.y==0
// stores W to rhs2[k]; others update rhs[i] -= L(i,k) @ W  (i = k + blockIdx.y).
__global__ void fwd_step(const float* __restrict__ lhs, const float* __restrict__ invLU,
                         float* __restrict__ rhs, float* __restrict__ rhs2, int k) {
  __shared__ float Wl[256];
  int lane = threadIdx.x;
  int jt = blockIdx.x;
  const float* invL = invLU + (size_t)k * 512;
  const float* B = rhs + (size_t)(k * 16) * S_N + jt * 16;
  v8f w;
#pragma unroll
  for (int v = 0; v < 8; ++v) w[v] = 0.f;
#pragma unroll
  for (int kb = 0; kb < 16; kb += 4) {
    v2f a = load_a(invL, 16, kb, lane);
    v2f b = load_b(B, S_N, kb, lane);
    w = __builtin_amdgcn_wmma_f32_16x16x4_f32(false, a, false, b, (short)0, w, false, false);
  }
  store_c(Wl, 16, lane, w);
  __syncthreads();
  if (blockIdx.y == 0) {
    store_c(rhs2 + (size_t)(k * 16) * S_N + jt * 16, S_N, lane, w);
  } else {
    int i = k + (int)blockIdx.y;
    const float* A = lhs + (size_t)(i * 16) * NN + k * 16;
    float* C = rhs + (size_t)(i * 16) * S_N + jt * 16;
    v8f c = load_c(C, S_N, lane);
#pragma unroll
    for (int kb = 0; kb < 16; kb += 4) {
      v2f a = load_a_neg(A, NN, kb, lane);
      v2f b = load_b(Wl, 16, kb, lane);
      c = __builtin_amdgcn_wmma_f32_16x16x4_f32(false, a, false, b, (short)0, c, false, false);
    }
    store_c(C, S_N, lane, c);
  }
}

// Back: W = invU_kk @ rhs2[k]; blockIdx.y==0 stores final coeffs to rhs[k];
// others update rhs2[i] -= U(i,k) @ W  (i = blockIdx.y - 1 < k).
__global__ void back_step(const float* __restrict__ lhs, const float* __restrict__ invLU,
                          float* __restrict__ rhs, float* __restrict__ rhs2, int k) {
  __shared__ float Wl[256];
  int lane = threadIdx.x;
  int jt = blockIdx.x;
  const float* invU = invLU + (size_t)k * 512 + 256;
  const float* B = rhs2 + (size_t)(k * 16) * S_N + jt * 16;
  v8f w;
#pragma unroll
  for (int v = 0; v < 8; ++v) w[v] = 0.f;
#pragma unroll
  for (int kb = 0; kb < 16; kb += 4) {
    v2f a = load_a(invU, 16, kb, lane);
    v2f b = load_b(B, S_N, kb, lane);
    w = __builtin_amdgcn_wmma_f32_16x16x4_f32(false, a, false, b, (short)0, w, false, false);
  }
  store_c(Wl, 16, lane, w);
  __syncthreads();
  if (blockIdx.y == 0) {
    store_c(rhs + (size_t)(k * 16) * S_N + jt * 16, S_N, lane, w);
  } else {
    int i = (int)blockIdx.y - 1;
    const float* A = lhs + (size_t)(i * 16) * NN + k * 16;
    float* C = rhs2 + (size_t)(i * 16) * S_N + jt * 16;
    v8f c = load_c(C, S_N, lane);
#pragma unroll
    for (int kb = 0; kb < 16; kb += 4) {
      v2f a = load_a_neg(A, NN, kb, lane);
      v2f b = load_b(Wl, 16, kb, lane);
      c = __builtin_amdgcn_wmma_f32_16x16x4_f32(false, a, false, b, (short)0, c, false, false);
    }
    store_c(C, S_N, lane, c);
  }
}

// ---------------- fused evaluation: out = [tps(x,y) | polyx] @ coeffs ----------------

__global__ void eval_out(const float* __restrict__ x, const float* __restrict__ y,
                         const float* __restrict__ stats, const float* __restrict__ coeffs,
                         float* __restrict__ out) {
  int lane = threadIdx.x;
  int qt = blockIdx.y;   // 0..Q_N/16-1
  int jt = blockIdx.x;   // 0..3
  int row = lane & 15;
  int q = qt * 16 + row;
  float x0 = x[q * 3 + 0], x1 = x[q * 3 + 1], x2 = x[q * 3 + 2];
  int col = jt * 16 + (lane & 15);
  int koff = (lane >> 4) << 1;

  v8f c;
#pragma unroll
  for (int v = 0; v < 8; ++v) c[v] = 0.f;

  for (int pt = 0; pt < P_N / 16; ++pt) {
    int pbase = pt * 16;
    __builtin_prefetch(&coeffs[(size_t)(pbase + 16 + koff) * S_N + col], 0, 0);
#pragma unroll
    for (int kb = 0; kb < 16; kb += 4) {
      int kk = pbase + kb + koff;
      float d0x = x0 - y[kk * 3 + 0];
      float d0y = x1 - y[kk * 3 + 1];
      float d0z = x2 - y[kk * 3 + 2];
      float d1x = x0 - y[(kk + 1) * 3 + 0];
      float d1y = x1 - y[(kk + 1) * 3 + 1];
      float d1z = x2 - y[(kk + 1) * 3 + 2];
      v2f a;
      a.x = tps(d0x * d0x + d0y * d0y + d0z * d0z);
      a.y = tps(d1x * d1x + d1y * d1y + d1z * d1z);
      v2f b;
      b.x = coeffs[(size_t)kk * S_N + col];
      b.y = coeffs[(size_t)(kk + 1) * S_N + col];
      c = __builtin_amdgcn_wmma_f32_16x16x4_f32(false, a, false, b, (short)0, c, false, false);
    }
  }
  // polynomial tail: rows P..P+15 of vec are [1, xhat0, xhat1, xhat2, 0...0]
  float xh0 = (x0 - stats[0]) / stats[3];
  float xh1 = (x1 - stats[1]) / stats[4];
  float xh2 = (x2 - stats[2]) / stats[5];
#pragma unroll
  for (int kb = 0; kb < 16; kb += 4) {
    int kkL = kb + koff;
    v2f a;
    a.x = (kkL == 0) ? 1.f : ((kkL == 1) ? xh0 : (kkL == 2) ? xh1 : (kkL == 3) ? xh2 : 0.f);
    int kL1 = kkL + 1;
    a.y = (kL1 == 1) ? xh0 : (kL1 == 2) ? xh1 : (kL1 == 3) ? xh2 : 0.f;
    v2f b;
    b.x = coeffs[(size_t)(P_N + kkL) * S_N + col];
    b.y = coeffs[(size_t)(P_N + kL1) * S_N + col];
    c = __builtin_amdgcn_wmma_f32_16x16x4_f32(false, a, false, b, (short)0, c, false, false);
  }
  store_c(out + (size_t)(qt * 16) * S_N + jt * 16, S_N, lane, c);
}

// ---------------- host launch ----------------

extern "C" void kernel_launch(void* const* d_in, const int* in_sizes, int n_in,
                              void* d_out, int out_size, void* d_ws, size_t ws_size,
                              hipStream_t stream) {
  (void)in_sizes; (void)n_in; (void)out_size; (void)ws_size;
  const float* x = (const float*)d_in[0];  // (8192,3)
  const float* y = (const float*)d_in[1];  // (4096,3)
  const float* d = (const float*)d_in[2];  // (4096,64)
  float* out = (float*)d_out;              // (8192,64)

  float* lhs   = (float*)d_ws;                       // NN*NN
  float* rhs   = lhs + (size_t)NN * NN;              // NN*S_N
  float* rhs2  = rhs + (size_t)NN * S_N;             // NN*S_N
  float* invLU = rhs2 + (size_t)NN * S_N;            // K16*512
  float* stats = invLU + (size_t)K16 * 512;          // 8 floats
  int*   ipiv  = (int*)(stats + 8);                  // NN ints

  // 1) y stats (shift/scale)
  stats_kernel<<<1, 256, 0, stream>>>(y, stats);
  // 2) build padded system
  build_lhs<<<dim3(K16, K16), 256, 0, stream>>>(y, stats, lhs);
  build_rhs<<<(NN * S_N + 255) / 256, 256, 0, stream>>>(d, rhs);

  // 3) blocked LU with partial pivoting
  for (int k = 0; k < K16; ++k) {
    panel_factor<<<1, 1024, 0, stream>>>(lhs, ipiv, invLU, k);
    int t = K16 - 1 - k;
    if (t > 0) {
      row_trsm<<<t, 32, 0, stream>>>(lhs, invLU, k);
      int g = (t + 1) / 2;
      lu_gemm<<<dim3(g, g), 32, 0, stream>>>(lhs, k);
    }
  }

  // 4) solve: coeffs = U^-1 L^-1 P rhs  (forward into rhs2, back into rhs)
  laswp_rhs<<<1, S_N, 0, stream>>>(rhs, ipiv);
  for (int k = 0; k < K16; ++k)
    fwd_step<<<dim3(S_N / 16, K16 - k), 32, 0, stream>>>(lhs, invLU, rhs, rhs2, k);
  for (int k = K16 - 1; k >= 0; --k)
    back_step<<<dim3(S_N / 16, k + 1), 32, 0, stream>>>(lhs, invLU, rhs, rhs2, k);

  // 5) fused evaluation GEMM (Kx computed on the fly)
  eval_out<<<dim3(S_N / 16, Q_N / 16), 32, 0, stream>>>(x, y, stats, rhs, out);
}